// BMPLoss_24670292148307
// MI455X (gfx1250) — compile-verified
//
#include <hip/hip_runtime.h>
#include <math.h>

#define B_SZ 512
#define J_SZ 24
#define V_SZ 6890
#define BATCH_ELEMS (V_SZ * 3)            // 20670 floats per batch
#define TOT_VERT (B_SZ * BATCH_ELEMS)     // 10,583,040
#define N4 (TOT_VERT / 4)                 // 2,645,760 float4 (exact)
#define BPW 5                             // batches packed per wave (5*3=15 <= 16 WMMA rows)

typedef __attribute__((ext_vector_type(2))) float v2f;
typedef __attribute__((ext_vector_type(8))) float v8f;

// ---- wave32 reduction (CDNA5 is wave32-only) ----
__device__ __forceinline__ float wave_sum(float v) {
#pragma unroll
  for (int off = 16; off > 0; off >>= 1) v += __shfl_down(v, off, 32);
  return v;
}

__device__ __forceinline__ void atomic_add_ws(float* p, float v) {
  unsafeAtomicAdd(p, v);  // -> global_atomic_add_f32
}

// ws layout: 0=shape_sum 1=n_valid 2=kp2d_sum 3=kp3d_sum 4=pose_sum 5=betas_sum 6=pa_sum
__global__ void init_ws(float* __restrict__ ws) {
  if (threadIdx.x < 16) ws[threadIdx.x] = 0.f;
}

// ---- dominant kernel: masked L1 over 2x42MB vertex streams (HBM-bound) ----
__global__ void vertex_loss(const float4* __restrict__ pv, const float4* __restrict__ gv,
                            const int* __restrict__ has_smpl, float* __restrict__ ws) {
  __shared__ float smask[B_SZ];
  for (int i = threadIdx.x; i < B_SZ; i += blockDim.x)
    smask[i] = (has_smpl[i] > 0) ? 1.f : 0.f;
  __syncthreads();

  int tid = blockIdx.x * blockDim.x + threadIdx.x;
  int stride = gridDim.x * blockDim.x;
  float lsum = 0.f;
  for (int i = tid; i < N4; i += stride) {
    // L2 lookahead for the pure stream (speculative; silently dropped if unmapped)
    __builtin_prefetch(&pv[i + stride], 0, 0);
    __builtin_prefetch(&gv[i + stride], 0, 0);
    float4 p = pv[i];
    float4 g = gv[i];
    int base = i * 4;  // batch stride 20670 is not a multiple of 4 -> per-element batch
    lsum += smask[(base    ) / BATCH_ELEMS] * fabsf(p.x - g.x)
          + smask[(base + 1) / BATCH_ELEMS] * fabsf(p.y - g.y)
          + smask[(base + 2) / BATCH_ELEMS] * fabsf(p.z - g.z)
          + smask[(base + 3) / BATCH_ELEMS] * fabsf(p.w - g.w);
  }
  lsum = wave_sum(lsum);
  if ((threadIdx.x & 31) == 0) atomic_add_ws(&ws[0], lsum);
}

// ---- keypoint losses: one thread per (b,j), exactly 512*24 = 12288 threads ----
__global__ void kp_loss(const float* __restrict__ cam, const float* __restrict__ pj,
                        const float* __restrict__ g2, const float* __restrict__ g3,
                        float* __restrict__ ws) {
  int i = blockIdx.x * blockDim.x + threadIdx.x;
  int b = i / J_SZ, j = i % J_SZ;
  const float f = 1000.f, img = 512.f;

  float c0 = cam[b * 3 + 0], c1 = cam[b * 3 + 1], c2 = cam[b * 3 + 2];
  float depth = 2.f * f / (1e-8f + c0 * img);
  float px = pj[b * 72 + j * 3 + 0] + c1;
  float py = pj[b * 72 + j * 3 + 1] + c2;
  float pz = pj[b * 72 + j * 3 + 2] + depth;
  float knx = (f * px / pz + img * 0.5f) / img;
  float kny = (f * py / pz + img * 0.5f) / img;
  float gx = g2[b * 72 + j * 3 + 0] / img;
  float gy = g2[b * 72 + j * 3 + 1] / img;
  float conf2 = g2[b * 72 + j * 3 + 2];
  float s2 = conf2 * (fabsf(knx - gx) + fabsf(kny - gy));

  float conf3 = g3[b * 96 + j * 4 + 3];
  float s3 = 0.f;
#pragma unroll
  for (int a = 0; a < 3; ++a) {
    float pp = (pj[b * 72 + 2 * 3 + a] + pj[b * 72 + 3 * 3 + a]) * 0.5f;  // pred pelvis
    float gp = (g3[b * 96 + 2 * 4 + a] + g3[b * 96 + 3 * 4 + a]) * 0.5f;  // gt pelvis
    float d = (pj[b * 72 + j * 3 + a] - pp) - (g3[b * 96 + j * 4 + a] - gp);
    s3 += fabsf(d);
  }
  s3 *= conf3;

  s2 = wave_sum(s2);
  s3 = wave_sum(s3);
  if ((threadIdx.x & 31) == 0) { atomic_add_ws(&ws[2], s2); atomic_add_ws(&ws[3], s3); }
}

// ---- pose loss: exactly 512*216 = 110592 threads ----
__global__ void pose_loss(const float* __restrict__ pr, const float* __restrict__ gr,
                          const int* __restrict__ has_smpl, float* __restrict__ ws) {
  int i = blockIdx.x * blockDim.x + threadIdx.x;
  int b = i / 216;
  float m = (has_smpl[b] > 0) ? 1.f : 0.f;
  float d = pr[i] - gr[i];
  float s = wave_sum(m * d * d);
  if ((threadIdx.x & 31) == 0) atomic_add_ws(&ws[4], s);
}

// ---- per-batch: n_valid + betas loss, exactly 512 threads ----
__global__ void batch_loss(const float* __restrict__ pb, const float* __restrict__ gs,
                           const int* __restrict__ has_smpl, float* __restrict__ ws) {
  int i = blockIdx.x * blockDim.x + threadIdx.x;
  float m = (has_smpl[i] > 0) ? 1.f : 0.f;
  float sb = 0.f;
#pragma unroll
  for (int k = 0; k < 10; ++k) { float d = pb[i * 10 + k] - gs[i * 10 + k]; sb += d * d; }
  sb *= m;
  float mv = wave_sum(m);
  sb = wave_sum(sb);
  if ((threadIdx.x & 31) == 0) { atomic_add_ws(&ws[1], mv); atomic_add_ws(&ws[5], sb); }
}

// ---- Procrustes: 5 batches per wave; K_b = X1*X2^T via block-diagonal
//      V_WMMA_F32_16X16X4_F32 (6 chained WMMAs, K accumulated in f32 C frag);
//      then 3x3 Jacobi-SVD per batch on one lane. One wave per block.
//      Row-padded [16][24] LDS layout -> branch-free fragment gather. ----
__global__ void procrustes_kernel(const float* __restrict__ pj, const float* __restrict__ g3,
                                  float* __restrict__ ws) {
  __shared__ float sP[BPW][24][3];
  __shared__ float sG[BPW][24][3];
  __shared__ float sX1r[16][24];      // row = 3*batch + axis; rows >= 3*nb and row 15 zeroed
  __shared__ float sX2r[16][24];
  __shared__ float sVarP[16];         // per-row partial var of X1
  __shared__ float sVar[BPW];
  __shared__ float sC[16][16];

  int t = threadIdx.x;              // 0..31 (one wave)
  int b0 = blockIdx.x * BPW;
  int nb = B_SZ - b0; if (nb > BPW) nb = BPW;

  // stage joints into LDS
  for (int i = t; i < nb * 72; i += 32) {
    int bl = i / 72, r = i % 72, j = r / 3, a = r % 3;
    sP[bl][j][a] = pj[(b0 + bl) * 72 + r];
    sG[bl][j][a] = g3[(b0 + bl) * 96 + j * 4 + a];
  }
  __syncthreads();

  // center; one lane per padded row (16 rows). Unused rows zero-filled so the
  // 16x16 WMMA tile stays clean and the fragment gather needs no guard.
  if (t < 16) {
    int row = t;
    int bl = row / 3, a = row % 3;
    if (row < 15 && bl < nb) {
      float m1 = 0.f, m2 = 0.f;
      for (int j = 0; j < 24; ++j) { m1 += sP[bl][j][a]; m2 += sG[bl][j][a]; }
      m1 *= (1.f / 24.f); m2 *= (1.f / 24.f);
      float vp = 0.f;
      for (int j = 0; j < 24; ++j) {
        float x1 = sP[bl][j][a] - m1;
        float x2 = sG[bl][j][a] - m2;
        sX1r[row][j] = x1; sX2r[row][j] = x2;
        vp += x1 * x1;
      }
      sVarP[row] = vp;
    } else {
      for (int j = 0; j < 24; ++j) { sX1r[row][j] = 0.f; sX2r[row][j] = 0.f; }
      sVarP[row] = 0.f;
    }
  }
  __syncthreads();
  if (t < BPW) sVar[t] = sVarP[3 * t + 0] + sVarP[3 * t + 1] + sVarP[3 * t + 2];
  __syncthreads();

  // ---- WMMA: C(16x16,f32) += A(16x4) * B(4x16), looped over 6 K-chunks of 4 joints.
  // A frag (ISA 7.12.2): lane l, vgpr v -> M = l&15, K = (l>=16?2:0)+v
  // B frag (C-like):     lane l, vgpr v -> N = l&15, K = (l>=16?2:0)+v
  // Diagonal 3x3 blocks of C are K_b for the 5 packed batches. EXEC all-1s,
  // fragment loads unconditional (consecutive j -> mergeable ds_load_b64).
  v8f c = {};
  int M = t & 15;
  int kh = (t >> 4) * 2;
#pragma unroll
  for (int kc = 0; kc < 6; ++kc) {
    int j0 = kc * 4 + kh;
    v2f av, bv;
    av[0] = sX1r[M][j0];
    av[1] = sX1r[M][j0 + 1];
    bv[0] = sX2r[M][j0];
    bv[1] = sX2r[M][j0 + 1];
    c = __builtin_amdgcn_wmma_f32_16x16x4_f32(false, av, false, bv, (short)0, c, false, false);
  }
  // scatter C frag to LDS: lane l, vgpr v -> row v + (l>=16?8:0), col l&15
  int rowBase = (t >> 4) * 8;
#pragma unroll
  for (int v = 0; v < 8; ++v) sC[rowBase + v][M] = c[v];
  __syncthreads();

  // ---- per-batch 3x3 SVD + Procrustes residual (one lane per batch) ----
  if (t < nb) {
    int bl = t;
    float Km[3][3];
#pragma unroll
    for (int r = 0; r < 3; ++r)
#pragma unroll
      for (int cc = 0; cc < 3; ++cc)
        Km[r][cc] = sC[3 * bl + r][3 * bl + cc] + 1e-8f;  // reference adds EPS to K

    // A = K^T K, Jacobi eigendecomposition A = V D V^T
    float Am[3][3];
#pragma unroll
    for (int r = 0; r < 3; ++r)
#pragma unroll
      for (int cc = 0; cc < 3; ++cc)
        Am[r][cc] = Km[0][r] * Km[0][cc] + Km[1][r] * Km[1][cc] + Km[2][r] * Km[2][cc];
    float Vm[3][3] = {{1.f, 0.f, 0.f}, {0.f, 1.f, 0.f}, {0.f, 0.f, 1.f}};
    const int PQ[3][2] = {{0, 1}, {0, 2}, {1, 2}};
    for (int sweep = 0; sweep < 8; ++sweep) {
      for (int r = 0; r < 3; ++r) {
        int p = PQ[r][0], q = PQ[r][1];
        float apq = Am[p][q];
        if (fabsf(apq) > 1e-30f) {
          float tau = (Am[q][q] - Am[p][p]) / (2.f * apq);
          float tt  = (tau >= 0.f ? 1.f : -1.f) / (fabsf(tau) + sqrtf(1.f + tau * tau));
          float cth = 1.f / sqrtf(1.f + tt * tt);
          float sth = tt * cth;
          for (int k = 0; k < 3; ++k) {
            float akp = Am[k][p], akq = Am[k][q];
            Am[k][p] = cth * akp - sth * akq;
            Am[k][q] = sth * akp + cth * akq;
          }
          for (int k = 0; k < 3; ++k) {
            float apk = Am[p][k], aqk = Am[q][k];
            Am[p][k] = cth * apk - sth * aqk;
            Am[q][k] = sth * apk + cth * aqk;
          }
          for (int k = 0; k < 3; ++k) {
            float vkp = Vm[k][p], vkq = Vm[k][q];
            Vm[k][p] = cth * vkp - sth * vkq;
            Vm[k][q] = sth * vkp + cth * vkq;
          }
        }
      }
    }
    float dv[3] = {Am[0][0], Am[1][1], Am[2][2]};
    // sort eigenvalues descending (svd order), permuting V columns
    for (int pass = 0; pass < 2; ++pass)
      for (int i2 = 0; i2 < 2; ++i2)
        if (dv[i2] < dv[i2 + 1]) {
          float td = dv[i2]; dv[i2] = dv[i2 + 1]; dv[i2 + 1] = td;
          for (int k = 0; k < 3; ++k) {
            float tv = Vm[k][i2]; Vm[k][i2] = Vm[k][i2 + 1]; Vm[k][i2 + 1] = tv;
          }
        }
    float sv[3];
#pragma unroll
    for (int i2 = 0; i2 < 3; ++i2) sv[i2] = sqrtf(fmaxf(dv[i2], 0.f));

    // sign(det(U V^T)) == sign(det(K)); Z negates smallest singular direction
    float detK = Km[0][0] * (Km[1][1] * Km[2][2] - Km[1][2] * Km[2][1])
               - Km[0][1] * (Km[1][0] * Km[2][2] - Km[1][2] * Km[2][0])
               + Km[0][2] * (Km[1][0] * Km[2][1] - Km[1][1] * Km[2][0]);
    float sgn = (detK >= 0.f) ? 1.f : -1.f;

    // U columns: u_i = K v_i / s_i
    float Um[3][3];
    for (int i2 = 0; i2 < 3; ++i2) {
      float inv = 1.f / fmaxf(sv[i2], 1e-20f);
      for (int r = 0; r < 3; ++r) {
        float w = Km[r][0] * Vm[0][i2] + Km[r][1] * Vm[1][i2] + Km[r][2] * Vm[2][i2];
        Um[r][i2] = w * inv;
      }
    }
    // R = V Z U^T  with Z = diag(1,1,sgn)
    float Rm[3][3];
    for (int r = 0; r < 3; ++r)
      for (int cc = 0; cc < 3; ++cc)
        Rm[r][cc] = Vm[r][0] * Um[cc][0] + Vm[r][1] * Um[cc][1] + sgn * Vm[r][2] * Um[cc][2];

    float scale = (sv[0] + sv[1] + sgn * sv[2]) / sVar[bl];  // trace(RK)/var1

    // S1_hat - S2 = scale*R*X1 - X2 (centered formulation)
    float acc = 0.f;
    for (int j = 0; j < 24; ++j) {
      float x0 = sX1r[3 * bl + 0][j], x1 = sX1r[3 * bl + 1][j], x2 = sX1r[3 * bl + 2][j];
      float e0 = scale * (Rm[0][0] * x0 + Rm[0][1] * x1 + Rm[0][2] * x2) - sX2r[3 * bl + 0][j];
      float e1 = scale * (Rm[1][0] * x0 + Rm[1][1] * x1 + Rm[1][2] * x2) - sX2r[3 * bl + 1][j];
      float e2 = scale * (Rm[2][0] * x0 + Rm[2][1] * x1 + Rm[2][2] * x2) - sX2r[3 * bl + 2][j];
      acc += sqrtf(e0 * e0 + e1 * e1 + e2 * e2);
    }
    atomic_add_ws(&ws[6], acc);
  }
}

// ---- combine (matches reference weights/denominators) ----
__global__ void finalize_kernel(const float* __restrict__ ws, float* __restrict__ out) {
  float nv = ws[1];
  float loss_shape = ws[0] / (nv * (float)(V_SZ * 3) + 1e-8f);
  float loss_pose  = ws[4] / (nv * 216.f + 1e-8f);
  float loss_betas = ws[5] / (nv * 10.f + 1e-8f);
  float loss_kp2d  = ws[2] / (float)(B_SZ * J_SZ * 2);
  float loss_kp3d  = ws[3] / (float)(B_SZ * J_SZ * 3);
  float pa         = ws[6] / (float)(B_SZ * J_SZ);
  out[0] = 4.f * loss_kp2d + 4.f * loss_kp3d + loss_shape + loss_pose + 0.01f * loss_betas + pa;
}

extern "C" void kernel_launch(void* const* d_in, const int* in_sizes, int n_in,
                              void* d_out, int out_size, void* d_ws, size_t ws_size,
                              hipStream_t stream) {
  const float* pred_rotmat   = (const float*)d_in[0];
  const float* pred_camera   = (const float*)d_in[1];
  const float* pred_joints   = (const float*)d_in[2];
  const float* pred_vertices = (const float*)d_in[3];
  const float* pred_betas    = (const float*)d_in[4];
  const float* gt_rotmat     = (const float*)d_in[5];
  const float* gt_shape      = (const float*)d_in[6];
  const float* gt_kp2d       = (const float*)d_in[7];
  const float* gt_kp3d       = (const float*)d_in[8];
  const float* gt_vertices   = (const float*)d_in[9];
  const int*   has_smpl      = (const int*)d_in[10];
  float* ws  = (float*)d_ws;
  float* out = (float*)d_out;

  init_ws<<<1, 32, 0, stream>>>(ws);
  vertex_loss<<<2048, 256, 0, stream>>>((const float4*)pred_vertices,
                                        (const float4*)gt_vertices, has_smpl, ws);
  kp_loss<<<48, 256, 0, stream>>>(pred_camera, pred_joints, gt_kp2d, gt_kp3d, ws);
  pose_loss<<<432, 256, 0, stream>>>(pred_rotmat, gt_rotmat, has_smpl, ws);
  batch_loss<<<2, 256, 0, stream>>>(pred_betas, gt_shape, has_smpl, ws);
  procrustes_kernel<<<(B_SZ + BPW - 1) / BPW, 32, 0, stream>>>(pred_joints, gt_kp3d, ws);
  finalize_kernel<<<1, 1, 0, stream>>>(ws, out);
}